// LLaRD_84731114816175
// MI455X (gfx1250) — compile-verified
//
#include <hip/hip_runtime.h>

#define NUM_D 64
#define TPB 256
#define WAVES_PER_BLOCK (TPB / 32)

// ---------------------------------------------------------------------------
// SpMM scatter: one wave32 per edge, 2 floats per lane.
// Source rows (256B each) of the next TWO edges are prefetched into LDS with
// global_load_async_to_lds_b64 (tracked by ASYNCcnt, depth-2 pipeline,
// triple-buffered) while the current edge's 64 global_atomic_add_f32
// scatters are issued.
// ---------------------------------------------------------------------------
__global__ __launch_bounds__(TPB) void lgcn_spmm(const int* __restrict__ rows,
                                                 const int* __restrict__ cols,
                                                 const float* __restrict__ vals,
                                                 const float* __restrict__ cur,
                                                 float* __restrict__ nxt,
                                                 int nnz) {
  __shared__ __align__(16) float stage[WAVES_PER_BLOCK][3][NUM_D];
  const int wib = threadIdx.x >> 5;  // wave in block
  const int lane = threadIdx.x & 31;
  const int wave = blockIdx.x * WAVES_PER_BLOCK + wib;
  const int nwaves = gridDim.x * WAVES_PER_BLOCK;

  // 32-bit LDS byte offsets of this lane's 8-byte slot in each of 3 buffers
  // (generic LDS address: low 32 bits are the LDS offset per the aperture map)
  unsigned lds[3];
  lds[0] = (unsigned)(uintptr_t)(&stage[wib][0][lane * 2]);
  lds[1] = (unsigned)(uintptr_t)(&stage[wib][1][lane * 2]);
  lds[2] = (unsigned)(uintptr_t)(&stage[wib][2][lane * 2]);

#define LGCN_ISSUE(EDGE, LDSADDR)                                              \
  do {                                                                         \
    unsigned long long _src = (unsigned long long)(uintptr_t)(                 \
        cur + (size_t)cols[(EDGE)] * NUM_D + lane * 2);                        \
    asm volatile("global_load_async_to_lds_b64 %0, %1, off" ::"v"(LDSADDR),    \
                 "v"(_src)                                                     \
                 : "memory");                                                  \
  } while (0)

  const int s = nwaves;
  int e = wave;
  if (e < nnz) LGCN_ISSUE(e, lds[0]);
  if (e + s < nnz) LGCN_ISSUE(e + s, lds[1]);

  int ib = 0;  // buffer holding the current edge's row
  for (; e < nnz; e += s) {
    const int e2 = e + 2 * s;
    int ibn = ib + 2;
    if (ibn >= 3) ibn -= 3;
    if (e2 < nnz) {
      LGCN_ISSUE(e2, lds[ibn]);
      // up to 3 outstanding after issue; oldest (current buffer) must be done
      asm volatile("s_wait_asynccnt 2" ::: "memory");
    } else if (e + s < nnz) {
      asm volatile("s_wait_asynccnt 1" ::: "memory");
    } else {
      asm volatile("s_wait_asynccnt 0" ::: "memory");
    }

    const float v = vals[e];  // wave-uniform
    const int r = rows[e];
    const float x0 = stage[wib][ib][lane * 2 + 0];
    const float x1 = stage[wib][ib][lane * 2 + 1];
    float* dst = nxt + (size_t)r * NUM_D + lane * 2;
    atomicAdd(dst + 0, v * x0);  // global_atomic_add_f32 (no return)
    atomicAdd(dst + 1, v * x1);

    ib = ib + 1;
    if (ib >= 3) ib = 0;
  }
#undef LGCN_ISSUE
}

// ---------------------------------------------------------------------------
// init: cur = ego = concat(user_emb, item_emb); acc = ego (layer-0 term);
//       zero the first scatter target in the same pass.
// ---------------------------------------------------------------------------
__global__ __launch_bounds__(TPB) void lgcn_init3(const float* __restrict__ ue,
                                                  const float* __restrict__ ie,
                                                  float* __restrict__ cur,
                                                  float* __restrict__ acc,
                                                  float* __restrict__ nxt,
                                                  long long userElems,
                                                  long long total) {
  long long i4 = (long long)blockIdx.x * TPB + threadIdx.x;
  long long i = i4 * 4;
  if (i >= total) return;
  float4 v = (i < userElems)
                 ? reinterpret_cast<const float4*>(ue)[i4]
                 : reinterpret_cast<const float4*>(ie)[(i - userElems) >> 2];
  reinterpret_cast<float4*>(cur)[i4] = v;
  reinterpret_cast<float4*>(acc)[i4] = v;
  reinterpret_cast<float4*>(nxt)[i4] = make_float4(0.f, 0.f, 0.f, 0.f);
}

// acc += nxt, and zero the buffer that becomes the next layer's scatter target
__global__ __launch_bounds__(TPB) void lgcn_accum_zero(
    float* __restrict__ acc, const float* __restrict__ nxt,
    float* __restrict__ zbuf, long long total) {
  long long i4 = (long long)blockIdx.x * TPB + threadIdx.x;
  if (i4 * 4 >= total) return;
  float4 a = reinterpret_cast<float4*>(acc)[i4];
  float4 b = reinterpret_cast<const float4*>(nxt)[i4];
  a.x += b.x; a.y += b.y; a.z += b.z; a.w += b.w;
  reinterpret_cast<float4*>(acc)[i4] = a;
  reinterpret_cast<float4*>(zbuf)[i4] = make_float4(0.f, 0.f, 0.f, 0.f);
}

// final: acc = (acc + nxt) * s   (fused last accumulate + mean scale)
__global__ __launch_bounds__(TPB) void lgcn_final(float* __restrict__ acc,
                                                  const float* __restrict__ nxt,
                                                  float s, long long total) {
  long long i4 = (long long)blockIdx.x * TPB + threadIdx.x;
  if (i4 * 4 >= total) return;
  float4 a = reinterpret_cast<float4*>(acc)[i4];
  float4 b = reinterpret_cast<const float4*>(nxt)[i4];
  a.x = (a.x + b.x) * s; a.y = (a.y + b.y) * s;
  a.z = (a.z + b.z) * s; a.w = (a.w + b.w) * s;
  reinterpret_cast<float4*>(acc)[i4] = a;
}

// ---------------------------------------------------------------------------
// Host-side orchestration (graph-capture safe: kernels on `stream` only)
// Inputs: [0]=adj_values f32[NNZ], [1]=user_emb f32[NU*64],
//         [2]=item_emb f32[NI*64], [3]=adj_indices i32[2*NNZ], [4]=gcn_layer
// Output: mean embeddings, user rows then item rows, flat [N*64] f32
// ---------------------------------------------------------------------------
extern "C" void kernel_launch(void* const* d_in, const int* in_sizes, int n_in,
                              void* d_out, int out_size, void* d_ws,
                              size_t ws_size, hipStream_t stream) {
  const float* adj_values = (const float*)d_in[0];
  const float* user_emb = (const float*)d_in[1];
  const float* item_emb = (const float*)d_in[2];
  const int* adj_indices = (const int*)d_in[3];

  const int nnz = in_sizes[0];
  const long long userElems = in_sizes[1];
  const long long itemElems = in_sizes[2];
  const long long total = userElems + itemElems;  // == out_size

  const int* rows = adj_indices;        // adj_indices[0, :]
  const int* cols = adj_indices + nnz;  // adj_indices[1, :]

  float* acc = (float*)d_out;
  float* bufA = (float*)d_ws;          // ws needs 2 * total * 4 B = 307.2 MB
  float* bufB = bufA + total;

  const long long total4 = total >> 2;
  const int ew_blocks = (int)((total4 + TPB - 1) / TPB);
  const int spmm_blocks = 4096;  // 32768 waves -> ~183 edges/wave pipeline
  const int GCN_LAYER = 3;       // fixed in the reference

  // cur = ego, acc = ego, first scatter target zeroed
  lgcn_init3<<<ew_blocks, TPB, 0, stream>>>(user_emb, item_emb, bufA, acc,
                                            bufB, userElems, total);
  float* cur = bufA;
  float* nxt = bufB;
  for (int l = 0; l < GCN_LAYER; ++l) {
    lgcn_spmm<<<spmm_blocks, TPB, 0, stream>>>(rows, cols, adj_values, cur, nxt,
                                               nnz);
    if (l == GCN_LAYER - 1) {
      // fused last accumulate + 1/(L+1) mean scale
      lgcn_final<<<ew_blocks, TPB, 0, stream>>>(acc, nxt,
                                                1.0f / (GCN_LAYER + 1), total);
    } else {
      // acc += nxt; zero the retired cur buffer (next layer's scatter target)
      lgcn_accum_zero<<<ew_blocks, TPB, 0, stream>>>(acc, nxt, cur, total);
    }
    float* t = cur; cur = nxt; nxt = t;
  }
}